// BoxMultiHeadedAttention_4776003633660
// MI455X (gfx1250) — compile-verified
//
#include <hip/hip_runtime.h>
#include <hip/hip_bf16.h>
#include <math.h>

#define NN 1024
#define DD 512
#define HH 8
#define DKK 64
#define DGG 64
#define LN_EPS 1e-5f

typedef float v2f __attribute__((ext_vector_type(2)));
typedef float v8f __attribute__((ext_vector_type(8)));

// ---------------------------------------------------------------------------
// Generic 32x32 output tile per wave using V_WMMA_F32_16X16X4_F32.
// Y[i][j] = sum_k Arow_i[k] * Brow_j[k]   (both operands row-contiguous in K)
//
// Fragment layout (ISA 7.12.2, 32-bit operands, wave32):
//   A 16x4: lanes 0-15 = rows M=0..15 holding K={k,k+1};
//           lanes 16-31 = same rows holding K={k+2,k+3}  -> contiguous v2f.
//   B 4x16: symmetric with N = lane%16.
//   C/D   : VGPR r -> M = r (lanes 0-15) / r+8 (lanes 16-31), N = lane%16.
//
// 2x2 accumulators -> 4 WMMA per (2 A + 2 B) b64 loads = 1.0 load/WMMA.
// ---------------------------------------------------------------------------
template <class Epi>
__device__ __forceinline__ void gemm_tile32(const float* __restrict__ A,
                                            int lda, const float* __restrict__ B,
                                            int ldb, int i0, int j0, int Ksize,
                                            Epi epi) {
  const int lane = threadIdx.x & 31;
  const int half = lane >> 4, lr = lane & 15;
  const float* ar0 = A + (size_t)(i0 + lr) * lda;
  const float* ar1 = ar0 + (size_t)16 * lda;
  const float* br0 = B + (size_t)(j0 + lr) * ldb;
  const float* br1 = br0 + (size_t)16 * ldb;

  v8f acc[2][2] = {};
#pragma unroll 8
  for (int k0 = 0; k0 < Ksize; k0 += 4) {
    const int kb = k0 + 2 * half;
    const v2f a0 = *(const v2f*)(ar0 + kb);
    const v2f a1 = *(const v2f*)(ar1 + kb);
    const v2f b0 = *(const v2f*)(br0 + kb);
    const v2f b1 = *(const v2f*)(br1 + kb);
    acc[0][0] = __builtin_amdgcn_wmma_f32_16x16x4_f32(false, a0, false, b0,
                                                      (short)0, acc[0][0], false, false);
    acc[0][1] = __builtin_amdgcn_wmma_f32_16x16x4_f32(false, a0, false, b1,
                                                      (short)0, acc[0][1], false, false);
    acc[1][0] = __builtin_amdgcn_wmma_f32_16x16x4_f32(false, a1, false, b0,
                                                      (short)0, acc[1][0], false, false);
    acc[1][1] = __builtin_amdgcn_wmma_f32_16x16x4_f32(false, a1, false, b1,
                                                      (short)0, acc[1][1], false, false);
  }

#pragma unroll
  for (int ti = 0; ti < 2; ++ti)
#pragma unroll
    for (int tj = 0; tj < 2; ++tj)
#pragma unroll
      for (int r = 0; r < 8; ++r)
        epi(i0 + ti * 16 + r + 8 * half, j0 + tj * 16 + lr, acc[ti][tj][r]);
}

// ---------------------------------------------------------------------------
// 1) Box-relative embedding + per-head gate, fused to log(clip(relu(.),1e-6)).
//    Writes S[h][i][j]; the scores kernel adds QK^T/8 in place afterwards.
// ---------------------------------------------------------------------------
__global__ void __launch_bounds__(256) gate_kernel(
    const float* __restrict__ box, const float* __restrict__ WGw,
    const float* __restrict__ WGb, float* __restrict__ S) {
  __shared__ float w_sh[HH * DGG];
  __shared__ float b_sh[HH];
  for (int t = threadIdx.x; t < HH * DGG; t += 256) w_sh[t] = WGw[t];
  if (threadIdx.x < HH) b_sh[threadIdx.x] = WGb[threadIdx.x];
  __syncthreads();

  const int i = blockIdx.y;
  const int j = blockIdx.x * 256 + threadIdx.x;

  const float xi0 = box[i * 4 + 0], yi0 = box[i * 4 + 1];
  const float xi1 = box[i * 4 + 2], yi1 = box[i * 4 + 3];
  const float cxi = (xi0 + xi1) * 0.5f, cyi = (yi0 + yi1) * 0.5f;
  const float wi = xi1 - xi0 + 1.0f, hi = yi1 - yi0 + 1.0f;

  const float xj0 = box[j * 4 + 0], yj0 = box[j * 4 + 1];
  const float xj1 = box[j * 4 + 2], yj1 = box[j * 4 + 3];
  const float cxj = (xj0 + xj1) * 0.5f, cyj = (yj0 + yj1) * 0.5f;
  const float wj = xj1 - xj0 + 1.0f, hj = yj1 - yj0 + 1.0f;

  float pos[4];
  pos[0] = __logf(fmaxf(fabsf((cxi - cxj) / wi), 0.001f));
  pos[1] = __logf(fmaxf(fabsf((cyi - cyj) / hi), 0.001f));
  pos[2] = __logf(wi / wj);
  pos[3] = __logf(hi / hj);

  const float dim_mat[8] = {1.0f, 0.42169650342858224f, 0.1778279410038923f,
                            0.07498942093324558f, 0.03162277660168379f,
                            0.013335214321633241f, 0.005623413251903491f,
                            0.0023713737056616554f};

  float acc[HH];
#pragma unroll
  for (int h = 0; h < HH; ++h) acc[h] = b_sh[h];

#pragma unroll
  for (int c = 0; c < 4; ++c) {
#pragma unroll
    for (int g = 0; g < 8; ++g) {
      const float m = 100.0f * pos[c] * dim_mat[g];
      const float sn = __sinf(m);
      const float cs = __cosf(m);
      const int f = c * 8 + g;
#pragma unroll
      for (int h = 0; h < HH; ++h)  // uniform-address LDS broadcast
        acc[h] += sn * w_sh[h * DGG + f] + cs * w_sh[h * DGG + 32 + f];
    }
  }
#pragma unroll
  for (int h = 0; h < HH; ++h)
    S[((size_t)h * NN + i) * NN + j] = __logf(fmaxf(acc[h], 1e-6f));
}

// ---------------------------------------------------------------------------
// 2) QKV projections: Y_z = X_z @ W_z^T + b_z.  32x32 tile/wave.
//    grid (2, 32, 3), block 256 (8 waves over j stripes).
// ---------------------------------------------------------------------------
__global__ void __launch_bounds__(256) qkv_kernel(
    const float* __restrict__ q_in, const float* __restrict__ k_in,
    const float* __restrict__ v_in, const float* __restrict__ Wq,
    const float* __restrict__ Wk, const float* __restrict__ Wv,
    const float* __restrict__ bq, const float* __restrict__ bk,
    const float* __restrict__ bv, float* __restrict__ qkv_out) {
  const int wave = threadIdx.x >> 5;
  const int z = blockIdx.z;
  const float* X = (z == 0) ? q_in : (z == 1) ? k_in : v_in;
  const float* W = (z == 0) ? Wq : (z == 1) ? Wk : Wv;
  const float* bias = (z == 0) ? bq : (z == 1) ? bk : bv;
  float* Y = qkv_out + (size_t)z * NN * DD;

  const int j0 = (blockIdx.x * 8 + wave) * 32;  // 16 j-stripes over D
  const int i0 = blockIdx.y * 32;               // 32 i-stripes over N

  gemm_tile32(X, DD, W, DD, i0, j0, DD, [&](int row, int col, float v) {
    Y[(size_t)row * DD + col] = v + bias[col];
  });
}

// ---------------------------------------------------------------------------
// 3) Scores: S[h] += (Q_h @ K_h^T) * 0.125, fused onto the gate-log buffer.
//    grid (4, 32, 8), block 256.
// ---------------------------------------------------------------------------
__global__ void __launch_bounds__(256) scores_kernel(
    const float* __restrict__ qkv, float* __restrict__ S) {
  const int wave = threadIdx.x >> 5;
  const int h = blockIdx.z;
  const int j0 = (blockIdx.x * 8 + wave) * 32;  // 32 j-stripes
  const int i0 = blockIdx.y * 32;               // 32 i-stripes

  const float* Q = qkv + h * DKK;
  const float* K = qkv + (size_t)NN * DD + h * DKK;
  float* Sh = S + (size_t)h * NN * NN;

  gemm_tile32(Q, DD, K, DD, i0, j0, DKK, [&](int row, int col, float v) {
    const size_t idx = (size_t)row * NN + col;
    Sh[idx] = Sh[idx] + v * 0.125f;
  });
}

// ---------------------------------------------------------------------------
// 4) Row softmax over 8192 rows of 1024 (in place).
// ---------------------------------------------------------------------------
__global__ void __launch_bounds__(256) softmax_kernel(float* __restrict__ S) {
  __shared__ float red[256];
  float* p = S + (size_t)blockIdx.x * NN;
  const int t = threadIdx.x;
  float vals[4];
  float m = -3.4e38f;
#pragma unroll
  for (int u = 0; u < 4; ++u) {
    vals[u] = p[t + 256 * u];
    m = fmaxf(m, vals[u]);
  }
  red[t] = m;
  __syncthreads();
  for (int s = 128; s > 0; s >>= 1) {
    if (t < s) red[t] = fmaxf(red[t], red[t + s]);
    __syncthreads();
  }
  m = red[0];
  __syncthreads();
  float sum = 0.0f;
#pragma unroll
  for (int u = 0; u < 4; ++u) {
    vals[u] = __expf(vals[u] - m);
    sum += vals[u];
  }
  red[t] = sum;
  __syncthreads();
  for (int s = 128; s > 0; s >>= 1) {
    if (t < s) red[t] += red[t + s];
    __syncthreads();
  }
  const float inv = 1.0f / red[0];
#pragma unroll
  for (int u = 0; u < 4; ++u) p[t + 256 * u] = vals[u] * inv;
}

// ---------------------------------------------------------------------------
// 5a) Transpose V into Vt[h][dk][n]  (flat: Vt[c][n] with c = h*64+dk),
//     so the AV GEMM's B operand is K-contiguous. 32x32 LDS tile transpose.
//     grid (32, 16), block 256 (32x8).
// ---------------------------------------------------------------------------
__global__ void __launch_bounds__(256) vtrans_kernel(
    const float* __restrict__ qkv, float* __restrict__ Vt) {
  __shared__ float tile[32][33];
  const float* V = qkv + (size_t)2 * NN * DD;
  const int n0 = blockIdx.x * 32, c0 = blockIdx.y * 32;
  const int tx = threadIdx.x & 31, ty = threadIdx.x >> 5;
#pragma unroll
  for (int r = 0; r < 32; r += 8)
    tile[ty + r][tx] = V[(size_t)(n0 + ty + r) * DD + c0 + tx];
  __syncthreads();
#pragma unroll
  for (int r = 0; r < 32; r += 8)
    Vt[(size_t)(c0 + ty + r) * NN + n0 + tx] = tile[tx][ty + r];
}

// ---------------------------------------------------------------------------
// 5b) Attention output: att[h] = W_mn @ V_h, B from Vt (row-contiguous).
//     grid (8, 8), block 256; wave -> one of 64 (i32 x j32) tiles per head.
// ---------------------------------------------------------------------------
__global__ void __launch_bounds__(256) av_kernel(
    const float* __restrict__ S, const float* __restrict__ Vt,
    float* __restrict__ att) {
  const int wave = threadIdx.x >> 5;
  const int h = blockIdx.y;
  const int tid = blockIdx.x * 8 + wave;  // 0..63
  const int i0 = (tid >> 1) * 32;         // 32 i-stripes
  const int j0 = (tid & 1) * 32;          // 2 j-stripes over DK

  const float* A = S + (size_t)h * NN * NN;
  const float* B = Vt + (size_t)h * DKK * NN;
  float* atth = att + (size_t)h * NN * DKK;

  gemm_tile32(A, NN, B, NN, i0, j0, NN, [&](int row, int col, float v) {
    atth[(size_t)row * DKK + col] = v;
  });
}

// ---------------------------------------------------------------------------
// 6) Faithful scramble (transpose(0,2,1).reshape) + residual + LayerNorm.
//    flat f = r*512+c maps to att[h = f>>16][n = f&1023][d = (f>>10)&63].
// ---------------------------------------------------------------------------
__global__ void __launch_bounds__(256) ln0_kernel(
    const float* __restrict__ att, const float* __restrict__ residual,
    const float* __restrict__ g, const float* __restrict__ b,
    float* __restrict__ out0) {
  __shared__ float red[256];
  const int r = blockIdx.x;
  const int t = threadIdx.x;
  float v[2];
#pragma unroll
  for (int u = 0; u < 2; ++u) {
    const int c = t + 256 * u;
    const unsigned f = (unsigned)r * 512u + (unsigned)c;
    const unsigned hh = f >> 16;
    const unsigned d = (f >> 10) & 63u;
    const unsigned n = f & 1023u;
    v[u] = att[((size_t)hh * NN + n) * DKK + d] + residual[(size_t)r * DD + c];
  }
  red[t] = v[0] + v[1];
  __syncthreads();
  for (int s = 128; s > 0; s >>= 1) {
    if (t < s) red[t] += red[t + s];
    __syncthreads();
  }
  const float mean = red[0] * (1.0f / DD);
  __syncthreads();
  const float d0 = v[0] - mean, d1 = v[1] - mean;
  red[t] = d0 * d0 + d1 * d1;
  __syncthreads();
  for (int s = 128; s > 0; s >>= 1) {
    if (t < s) red[t] += red[t + s];
    __syncthreads();
  }
  const float inv = rsqrtf(red[0] * (1.0f / DD) + LN_EPS);
#pragma unroll
  for (int u = 0; u < 2; ++u) {
    const int c = t + 256 * u;
    out0[(size_t)r * DD + c] = (u == 0 ? d0 : d1) * inv * g[c] + b[c];
  }
}

// ---------------------------------------------------------------------------
// 7) Output projection: proj = out0 @ Wo^T + bo.  grid (2, 32), block 256.
// ---------------------------------------------------------------------------
__global__ void __launch_bounds__(256) proj_kernel(
    const float* __restrict__ X, const float* __restrict__ W,
    const float* __restrict__ bias, float* __restrict__ Y) {
  const int wave = threadIdx.x >> 5;
  const int j0 = (blockIdx.x * 8 + wave) * 32;
  const int i0 = blockIdx.y * 32;
  gemm_tile32(X, DD, W, DD, i0, j0, DD, [&](int row, int col, float v) {
    Y[(size_t)row * DD + col] = v + bias[col];
  });
}

// ---------------------------------------------------------------------------
// 8) Final residual + LayerNorm -> d_out.
// ---------------------------------------------------------------------------
__global__ void __launch_bounds__(256) lnf_kernel(
    const float* __restrict__ proj, const float* __restrict__ out0,
    const float* __restrict__ g, const float* __restrict__ b,
    float* __restrict__ out) {
  __shared__ float red[256];
  const int r = blockIdx.x;
  const int t = threadIdx.x;
  float v[2];
#pragma unroll
  for (int u = 0; u < 2; ++u) {
    const int c = t + 256 * u;
    v[u] = proj[(size_t)r * DD + c] + out0[(size_t)r * DD + c];
  }
  red[t] = v[0] + v[1];
  __syncthreads();
  for (int s = 128; s > 0; s >>= 1) {
    if (t < s) red[t] += red[t + s];
    __syncthreads();
  }
  const float mean = red[0] * (1.0f / DD);
  __syncthreads();
  const float d0 = v[0] - mean, d1 = v[1] - mean;
  red[t] = d0 * d0 + d1 * d1;
  __syncthreads();
  for (int s = 128; s > 0; s >>= 1) {
    if (t < s) red[t] += red[t + s];
    __syncthreads();
  }
  const float inv = rsqrtf(red[0] * (1.0f / DD) + LN_EPS);
#pragma unroll
  for (int u = 0; u < 2; ++u) {
    const int c = t + 256 * u;
    out[(size_t)r * DD + c] = (u == 0 ? d0 : d1) * inv * g[c] + b[c];
  }
}

extern "C" void kernel_launch(void* const* d_in, const int* in_sizes, int n_in,
                              void* d_out, int out_size, void* d_ws,
                              size_t ws_size, hipStream_t stream) {
  (void)in_sizes; (void)n_in; (void)out_size; (void)ws_size;
  const float* q_in = (const float*)d_in[0];
  const float* k_in = (const float*)d_in[1];
  const float* v_in = (const float*)d_in[2];
  const float* box  = (const float*)d_in[3];
  const float* Wq = (const float*)d_in[4];
  const float* bq = (const float*)d_in[5];
  const float* Wk = (const float*)d_in[6];
  const float* bk = (const float*)d_in[7];
  const float* Wv = (const float*)d_in[8];
  const float* bv = (const float*)d_in[9];
  const float* Wo = (const float*)d_in[10];
  const float* bo = (const float*)d_in[11];
  const float* WGw = (const float*)d_in[12];
  const float* WGb = (const float*)d_in[13];
  const float* ln0g = (const float*)d_in[14];
  const float* ln0b = (const float*)d_in[15];
  const float* lng = (const float*)d_in[16];
  const float* lnb = (const float*)d_in[17];

  float* ws = (float*)d_ws;
  float* S    = ws;                          // 8*1024*1024    (32 MB)
  float* qkv  = S + (size_t)HH * NN * NN;    // 3*1024*512     ( 6 MB)
  float* Vt   = qkv + (size_t)3 * NN * DD;   // 8*64*1024      ( 2 MB)
  float* att  = Vt + (size_t)HH * DKK * NN;  // 8*1024*64      ( 2 MB)
  float* out0 = att + (size_t)HH * NN * DKK; // 1024*512       ( 2 MB)
  float* proj = out0 + (size_t)NN * DD;      // 1024*512       ( 2 MB)
  float* out  = (float*)d_out;

  gate_kernel<<<dim3(4, NN), 256, 0, stream>>>(box, WGw, WGb, S);
  qkv_kernel<<<dim3(2, 32, 3), 256, 0, stream>>>(q_in, k_in, v_in, Wq, Wk, Wv,
                                                 bq, bk, bv, qkv);
  scores_kernel<<<dim3(4, 32, 8), 256, 0, stream>>>(qkv, S);
  softmax_kernel<<<dim3(HH * NN), 256, 0, stream>>>(S);
  vtrans_kernel<<<dim3(32, 16), 256, 0, stream>>>(qkv, Vt);
  av_kernel<<<dim3(8, 8), 256, 0, stream>>>(S, Vt, att);
  ln0_kernel<<<dim3(NN), 256, 0, stream>>>(att, q_in, ln0g, ln0b, out0);
  proj_kernel<<<dim3(2, 32), 256, 0, stream>>>(out0, Wo, bo, proj);
  lnf_kernel<<<dim3(NN), 256, 0, stream>>>(proj, out0, lng, lnb, out);
}